// ChamferDistance_36369783063040
// MI455X (gfx1250) — compile-verified
//
#include <hip/hip_runtime.h>

typedef float v2f __attribute__((ext_vector_type(2)));
typedef float v8f __attribute__((ext_vector_type(8)));

#define NPTS    16384
#define NSPLIT  4                       // column-space splits per pass
#define CHUNK   2048                    // Q points staged per LDS iteration (32 KB)
#define WAVES   8                       // waves per block (wave32)
#define BLOCK   (WAVES * 32)
#define ROWS_PER_BLOCK  (WAVES * 16)    // 128
#define ROWBLOCKS       (NPTS / ROWS_PER_BLOCK)        // 128
#define BLOCKS_PER_PASS (NSPLIT * ROWBLOCKS)           // 512
#define COLS_PER_SPLIT  (NPTS / NSPLIT)                // 4096

// ws layout: [2 * NSPLIT * NPTS] row-min floats, then [128] block partials.
#define RM_FLOATS (2 * NSPLIT * NPTS)

// D = A x B + C with:
//   A[m][0..2] = -2*p_m, A[m][3] = 1          (loop-invariant, 2 VGPRs)
//   B[0..2][n] = q_n,    B[3][n] = |q_n|^2    (one ds_load_b64 per tile, pre-swizzled)
//   C[m][n]    = |p_m|^2                       (loop-invariant, 8 VGPRs)
// => D[m][n] = squared distance tile in a single v_wmma_f32_16x16x4_f32.
__global__ __launch_bounds__(BLOCK) void chamfer_pass(const float* __restrict__ A,
                                                      const float* __restrict__ B,
                                                      float* __restrict__ rowmins)
{
    // Pre-swizzled B operands: bops[t*32 + lane]:
    //   lanes 0-15 : (q.x, q.y)   -> B rows K=0,1
    //   lanes 16-31: (q.z, |q|^2) -> B rows K=2,3
    __shared__ v2f bops[2 * CHUNK];

    const int tid    = threadIdx.x;
    const int wave   = tid >> 5;
    const int lane   = tid & 31;
    const int lane16 = lane & 15;
    const int hi     = lane >> 4;

    const int bid      = blockIdx.x;
    const int pass     = bid / BLOCKS_PER_PASS;     // 0: min over b per a; 1: swapped
    const int lb       = bid % BLOCKS_PER_PASS;
    const int splitIdx = lb / ROWBLOCKS;
    const int rowBlock = lb % ROWBLOCKS;

    const float* __restrict__ P = pass ? B : A;
    const float* __restrict__ Q = pass ? A : B;

    const int m = rowBlock * ROWS_PER_BLOCK + wave * 16 + lane16;

    const float ax = P[m * 3 + 0];
    const float ay = P[m * 3 + 1];
    const float az = P[m * 3 + 2];
    const float a2 = ax * ax + ay * ay + az * az;

    // A operand (32-bit A 16x4): lanes 0-15 hold K=0,1; lanes 16-31 hold K=2,3.
    v2f aOp;
    aOp.x = hi ? (-2.0f * az) : (-2.0f * ax);
    aOp.y = hi ? 1.0f         : (-2.0f * ay);

    // C operand: element (lane, v) is row M = v + 8*hi -> a2 lives in lane (v + 8*hi).
    v8f cm;
    for (int v = 0; v < 8; ++v) cm[v] = __shfl(a2, (hi << 3) + v, 32);

    v8f rm;
    for (int v = 0; v < 8; ++v) rm[v] = 3.0e38f;

    const int colBase = splitIdx * COLS_PER_SPLIT;

    for (int c = 0; c < COLS_PER_SPLIT / CHUNK; ++c) {
        // Stage CHUNK Q points, pre-formatted as per-lane B operands.
        for (int i = tid; i < CHUNK; i += BLOCK) {
            const int g = colBase + c * CHUNK + i;
            const float qx = Q[g * 3 + 0];
            const float qy = Q[g * 3 + 1];
            const float qz = Q[g * 3 + 2];
            const int t   = i >> 4;
            const int col = i & 15;
            v2f lo;  lo.x  = qx; lo.y  = qy;
            v2f hi2; hi2.x = qz; hi2.y = qx * qx + qy * qy + qz * qz;
            bops[t * 32 + col]      = lo;
            bops[t * 32 + 16 + col] = hi2;
        }
        __syncthreads();

        for (int t = 0; t < CHUNK / 16; ++t) {
            const v2f bOp = bops[t * 32 + lane];          // single ds_load_b64
            v8f d = __builtin_amdgcn_wmma_f32_16x16x4_f32(
                false, aOp, false, bOp, (short)0, cm, false, false);
            for (int v = 0; v < 8; ++v) rm[v] = fminf(rm[v], d[v]);
        }
        __syncthreads();
    }

    // Row-min within the wave: element (M = v + 8*hi, N = lane&15);
    // xor-reduce across each 16-lane half -> all lanes hold the 8 row-mins of their half.
    for (int v = 0; v < 8; ++v) {
        float x = rm[v];
        x = fminf(x, __shfl_xor(x, 1, 32));
        x = fminf(x, __shfl_xor(x, 2, 32));
        x = fminf(x, __shfl_xor(x, 4, 32));
        x = fminf(x, __shfl_xor(x, 8, 32));
        rm[v] = x;
    }

    // Lanes 0 and 16 publish rows [hi*8, hi*8+8) of this wave's tile for this split.
    if (lane16 == 0) {
        const int rowsBase = (pass * NSPLIT + splitIdx) * NPTS
                           + rowBlock * ROWS_PER_BLOCK + wave * 16 + hi * 8;
        float4* dst = (float4*)&rowmins[rowsBase];
        dst[0] = make_float4(rm[0], rm[1], rm[2], rm[3]);
        dst[1] = make_float4(rm[4], rm[5], rm[6], rm[7]);
    }
}

// One thread per row (2*NPTS rows): min across NSPLIT splits, sqrt, block tree-sum.
__global__ __launch_bounds__(256) void chamfer_finalize1(const float* __restrict__ rowmins,
                                                         float* __restrict__ blockPartials)
{
    __shared__ float sm[256];
    const int g    = blockIdx.x * 256 + threadIdx.x;   // 0 .. 2*NPTS-1
    const int pass = g >> 14;                          // NPTS == 2^14
    const int row  = g & (NPTS - 1);

    float x = rowmins[(pass * NSPLIT + 0) * NPTS + row];
    for (int s = 1; s < NSPLIT; ++s)
        x = fminf(x, rowmins[(pass * NSPLIT + s) * NPTS + row]);

    sm[threadIdx.x] = sqrtf(fmaxf(x, 0.0f));
    __syncthreads();
    for (int o = 128; o > 0; o >>= 1) {
        if (threadIdx.x < o) sm[threadIdx.x] += sm[threadIdx.x + o];
        __syncthreads();
    }
    if (threadIdx.x == 0) blockPartials[blockIdx.x] = sm[0];
}

__global__ __launch_bounds__(128) void chamfer_finalize2(const float* __restrict__ blockPartials,
                                                         float* __restrict__ out)
{
    __shared__ float sm[128];
    sm[threadIdx.x] = blockPartials[threadIdx.x];
    __syncthreads();
    for (int o = 64; o > 0; o >>= 1) {
        if (threadIdx.x < o) sm[threadIdx.x] += sm[threadIdx.x + o];
        __syncthreads();
    }
    if (threadIdx.x == 0) out[0] = sm[0] / (float)(2 * NPTS);
}

extern "C" void kernel_launch(void* const* d_in, const int* in_sizes, int n_in,
                              void* d_out, int out_size, void* d_ws, size_t ws_size,
                              hipStream_t stream) {
    const float* a = (const float*)d_in[0];
    const float* b = (const float*)d_in[1];
    float* out     = (float*)d_out;
    float* rowmins = (float*)d_ws;                 // 2*NSPLIT*NPTS floats (512 KB)
    float* bp      = rowmins + RM_FLOATS;          // 128 block partials

    const int nRowBlocks = 2 * NPTS / 256;         // 128 finalize1 blocks

    chamfer_pass<<<2 * BLOCKS_PER_PASS, BLOCK, 0, stream>>>(a, b, rowmins);  // 1024 blocks
    chamfer_finalize1<<<nRowBlocks, 256, 0, stream>>>(rowmins, bp);
    chamfer_finalize2<<<1, 128, 0, stream>>>(bp, out);
}